// CrossDomainGAT_49606872269032
// MI455X (gfx1250) — compile-verified
//
#include <hip/hip_runtime.h>
#include <hip/hip_bf16.h>

// CrossDomainGAT for MI455X (gfx1250, wave32).
// N=50000, D=128, H=8, HD=16, E=800000.
//
// Pipeline:
//   k0: zero accumulator in workspace
//   k1: Q,K,V = x @ {Wq,Wk,Wv}^T         -- V_WMMA_F32_16X16X4_F32 (fp32 matrix pipe)
//   k2: per-edge attention + scatter-add  -- one wave32 per edge, global_atomic_add_f32
//   k3: out = acc @ Wo^T + bo + x, LayerNorm -- WMMA + in-LDS row reductions

typedef float v2f __attribute__((ext_vector_type(2)));
typedef float v8f __attribute__((ext_vector_type(8)));

#define D_DIM 128
#define ALPHA 0.2f
#define LN_EPS 1e-5f

// ---------------------------------------------------------------- zero scratch
__global__ void gat_zero_f32(float* __restrict__ p, size_t n) {
  size_t i = (size_t)blockIdx.x * blockDim.x + threadIdx.x;
  if (i < n) p[i] = 0.0f;
}

// ------------------------------------------------------------- QKV projections
// Block = 256 threads = 8 waves. Block handles a 16-row tile of x.
// Wave w computes the 16x16 output tile at columns [16w, 16w+16) for Q, K, V,
// reusing one A fragment (x) per K-step for all three weight matrices.
__global__ void __launch_bounds__(256)
gat_qkv_wmma(const float* __restrict__ x,
             const float* __restrict__ Wq,
             const float* __restrict__ Wk,
             const float* __restrict__ Wv,
             float* __restrict__ Q,
             float* __restrict__ K,
             float* __restrict__ V,
             int N) {
  const int lane  = threadIdx.x & 31;
  const int wave  = threadIdx.x >> 5;          // 0..7 -> N-tile
  const int row0  = blockIdx.x * 16;
  const int n0    = wave * 16;
  const int m     = lane & 15;                 // M for A-frag, N for B-frag
  const int khalf = (lane >> 4) * 2;           // 0 or 2

  int arow = row0 + m; if (arow >= N) arow = N - 1;   // clamp (EXEC must stay full)
  const float* xr  = x  + (size_t)arow * D_DIM;
  const float* wqr = Wq + (size_t)(n0 + m) * D_DIM;
  const float* wkr = Wk + (size_t)(n0 + m) * D_DIM;
  const float* wvr = Wv + (size_t)(n0 + m) * D_DIM;

  v8f cq = {}; v8f ck = {}; v8f cv = {};
  for (int k = 0; k < D_DIM; k += 4) {
    __builtin_prefetch(xr + k + 32, 0, 1);
    v2f a, bq, bk, bv;
    a.x  = xr [k + khalf];  a.y  = xr [k + khalf + 1];
    bq.x = wqr[k + khalf];  bq.y = wqr[k + khalf + 1];
    bk.x = wkr[k + khalf];  bk.y = wkr[k + khalf + 1];
    bv.x = wvr[k + khalf];  bv.y = wvr[k + khalf + 1];
    cq = __builtin_amdgcn_wmma_f32_16x16x4_f32(false, a, false, bq, (short)0, cq, false, false);
    ck = __builtin_amdgcn_wmma_f32_16x16x4_f32(false, a, false, bk, (short)0, ck, false, false);
    cv = __builtin_amdgcn_wmma_f32_16x16x4_f32(false, a, false, bv, (short)0, cv, false, false);
  }

  const int rbase = (lane >> 4) << 3;          // 0 or 8
  const int c0    = n0 + (lane & 15);
#pragma unroll
  for (int v = 0; v < 8; ++v) {
    int r = row0 + v + rbase;
    if (r < N) {
      size_t o = (size_t)r * D_DIM + c0;
      Q[o] = cq[v];
      K[o] = ck[v];
      V[o] = cv[v];
    }
  }
}

// --------------------------------------------------------------- edge attention
// One wave32 per edge. Lane covers elements {lane, lane+32, lane+64, lane+96};
// element (lane + 32j) belongs to head h = 2j + (lane>=16).
__global__ void __launch_bounds__(256)
gat_edge_attn(const long long* __restrict__ ei,   // [2, E] int64
              const float* __restrict__ ea,       // [E, 16]
              const float* __restrict__ Q,
              const float* __restrict__ K,
              const float* __restrict__ V,
              float* __restrict__ acc,            // [N, 128]
              long long E) {
  const int lane = threadIdx.x & 31;
  const long long e = (long long)blockIdx.x * 8 + (threadIdx.x >> 5);
  if (e >= E) return;

  const int row = (int)ei[e];
  const int col = (int)ei[E + e];
  const float* qr = Q + (size_t)row * D_DIM;
  const float* kc = K + (size_t)col * D_DIM;
  const float* vr = V + (size_t)row * D_DIM;

  // partial per-head dot products
  float p[4];
#pragma unroll
  for (int j = 0; j < 4; ++j) {
    int idx = lane + 32 * j;
    p[j] = qr[idx] * kc[idx];
  }
  // butterfly reduce within each 16-lane half (16 elements of one head)
#pragma unroll
  for (int off = 1; off < 16; off <<= 1)
#pragma unroll
    for (int j = 0; j < 4; ++j)
      p[j] += __shfl_xor(p[j], off, 32);

  // other half's four head-sums
  float q4[4];
#pragma unroll
  for (int j = 0; j < 4; ++j) q4[j] = __shfl_xor(p[j], 16, 32);

  const bool hi = (lane >= 16);
  float s[8];
#pragma unroll
  for (int j = 0; j < 4; ++j) {
    s[2 * j]     = hi ? q4[j] : p[j];   // head 2j lives in low half
    s[2 * j + 1] = hi ? p[j]  : q4[j];  // head 2j+1 lives in high half
  }

  // edge gate: sigmoid(sum of edge_attr[e, 0..15])
  float t = (lane < 16) ? ea[(size_t)e * 16 + lane] : 0.0f;
#pragma unroll
  for (int off = 1; off < 32; off <<= 1) t += __shfl_xor(t, off, 32);
  const float ew = 1.0f / (1.0f + __expf(-t));

  // scale, LeakyReLU, gate
  const float inv_sqrt_hd = 0.25f;  // 1/sqrt(16)
#pragma unroll
  for (int h = 0; h < 8; ++h) {
    float v = s[h] * inv_sqrt_hd;
    v = (v > 0.0f) ? v : ALPHA * v;
    s[h] = v * ew;
  }

  // softmax over heads
  float mx = s[0];
#pragma unroll
  for (int h = 1; h < 8; ++h) mx = fmaxf(mx, s[h]);
  float sum = 0.0f;
#pragma unroll
  for (int h = 0; h < 8; ++h) { s[h] = __expf(s[h] - mx); sum += s[h]; }
  const float inv = 1.0f / sum;

  // weighted V, scatter-add into destination node
  float* ac = acc + (size_t)col * D_DIM;
#pragma unroll
  for (int j = 0; j < 4; ++j) {
    int idx = lane + 32 * j;
    float ph = (hi ? s[2 * j + 1] : s[2 * j]) * inv;
    atomicAdd(&ac[idx], ph * vr[idx]);
  }
}

// ---------------------------------------------- output GEMM + residual + LayerNorm
// Block = 256 threads = 8 waves computing one full 16x128 row tile; tile goes to
// LDS, then 16 threads per row do the LayerNorm reductions via half-wave shuffles.
__global__ void __launch_bounds__(256)
gat_out_ln(const float* __restrict__ acc,
           const float* __restrict__ Wo,
           const float* __restrict__ bo,
           const float* __restrict__ x,
           const float* __restrict__ gamma,
           const float* __restrict__ beta,
           float* __restrict__ out,
           int N) {
  __shared__ float tile[16][D_DIM + 4];

  const int lane  = threadIdx.x & 31;
  const int wave  = threadIdx.x >> 5;
  const int row0  = blockIdx.x * 16;
  const int n0    = wave * 16;
  const int m     = lane & 15;
  const int khalf = (lane >> 4) * 2;

  int arow = row0 + m; if (arow >= N) arow = N - 1;
  const float* ar = acc + (size_t)arow * D_DIM;
  const float* wr = Wo + (size_t)(n0 + m) * D_DIM;

  v8f c = {};
  for (int k = 0; k < D_DIM; k += 4) {
    v2f a, b;
    a.x = ar[k + khalf]; a.y = ar[k + khalf + 1];
    b.x = wr[k + khalf]; b.y = wr[k + khalf + 1];
    c = __builtin_amdgcn_wmma_f32_16x16x4_f32(false, a, false, b, (short)0, c, false, false);
  }

  const int rbase = (lane >> 4) << 3;
  const int c0    = n0 + (lane & 15);
  const float bias = bo[c0];
#pragma unroll
  for (int v = 0; v < 8; ++v) {
    int r = v + rbase;                        // 0..15 within tile
    int gr = row0 + r;
    float resid = (gr < N) ? x[(size_t)gr * D_DIM + c0] : 0.0f;
    tile[r][c0] = c[v] + bias + resid;        // GEMM + bias + residual
  }
  __syncthreads();

  // LayerNorm: 16 threads per row (threads of one 16-lane half-wave)
  const int row = threadIdx.x >> 4;           // 0..15
  const int sub = threadIdx.x & 15;
  float vals[8];
  float s = 0.0f;
#pragma unroll
  for (int j = 0; j < 8; ++j) {
    float v = tile[row][sub + 16 * j];
    vals[j] = v;
    s += v;
  }
#pragma unroll
  for (int off = 8; off >= 1; off >>= 1) s += __shfl_xor(s, off, 32);
  const float mu = s * (1.0f / 128.0f);

  float s2 = 0.0f;
#pragma unroll
  for (int j = 0; j < 8; ++j) { float d = vals[j] - mu; s2 += d * d; }
#pragma unroll
  for (int off = 8; off >= 1; off >>= 1) s2 += __shfl_xor(s2, off, 32);
  const float rstd = rsqrtf(s2 * (1.0f / 128.0f) + LN_EPS);

  const int grow = row0 + row;
  if (grow < N) {
#pragma unroll
    for (int j = 0; j < 8; ++j) {
      int colI = sub + 16 * j;
      out[(size_t)grow * D_DIM + colI] =
          (vals[j] - mu) * rstd * gamma[colI] + beta[colI];
    }
  }
}

// ------------------------------------------------------------------- launcher
extern "C" void kernel_launch(void* const* d_in, const int* in_sizes, int n_in,
                              void* d_out, int out_size, void* d_ws, size_t ws_size,
                              hipStream_t stream) {
  const float*     x     = (const float*)d_in[0];
  const long long* ei    = (const long long*)d_in[1];
  const float*     ea    = (const float*)d_in[2];
  const float*     Wq    = (const float*)d_in[3];
  const float*     Wk    = (const float*)d_in[4];
  const float*     Wv    = (const float*)d_in[5];
  const float*     Wo    = (const float*)d_in[6];
  const float*     bo    = (const float*)d_in[7];
  const float*     gamma = (const float*)d_in[8];
  const float*     beta  = (const float*)d_in[9];

  const int       N = in_sizes[0] / D_DIM;     // 50000
  const long long E = in_sizes[1] / 2;         // 800000

  const size_t mat = (size_t)N * D_DIM;
  float* Q   = (float*)d_ws;
  float* K   = Q + mat;
  float* V   = K + mat;
  float* acc = V + mat;

  // k0: zero the scatter accumulator (ws is poisoned by the harness)
  gat_zero_f32<<<(unsigned)((mat + 255) / 256), 256, 0, stream>>>(acc, mat);

  // k1: Q/K/V projections (fp32 WMMA)
  const int mtiles = (N + 15) / 16;
  gat_qkv_wmma<<<mtiles, 256, 0, stream>>>(x, Wq, Wk, Wv, Q, K, V, N);

  // k2: per-edge attention + scatter-add (one wave per edge, 8 waves/block)
  const unsigned eblocks = (unsigned)((E + 7) / 8);
  gat_edge_attn<<<eblocks, 256, 0, stream>>>(ei, ea, Q, K, V, acc, E);

  // k3: output GEMM + bias + residual + LayerNorm
  gat_out_ln<<<mtiles, 256, 0, stream>>>(acc, Wo, bo, x, gamma, beta,
                                         (float*)d_out, N);
}